// MultiHeadAttentionQuantum_65481071399122
// MI455X (gfx1250) — compile-verified
//
#include <hip/hip_runtime.h>
#include <math.h>

// ---------------------------------------------------------------------------
// Quantum multi-head attention, MI455X (gfx1250) fused implementation.
//
// Closed form: qh[h][0]   = prod_{b=1..7} cos(x[h*8+b] + theta[b])
//              qh[h][w>0] = prod_{b=0..w} cos(x[h*8+b] + theta[b])
// (derived from the GF(2)-linear CNOT permutation; the 256-wide tensor
//  product + sign contraction collapses to prefix products of cos(x+theta)).
//
// Kernel 1: per-token heads + 16x16 attention + softmax + permuted write of Y.
// Kernel 2: Y(8192x128) @ W^T(128x128) + b via V_WMMA_F32_16X16X4_F32.
// ---------------------------------------------------------------------------

typedef float v2f __attribute__((ext_vector_type(2)));
typedef float v8f __attribute__((ext_vector_type(8)));

#define NW 8
#define NH 16
#define EDIM 128
#define TOK_PER_BLOCK 16

__global__ __launch_bounds__(256) void quantum_attn_kernel(
    const float* __restrict__ x,
    const float* __restrict__ theta,
    float* __restrict__ Y)
{
    __shared__ float qh_lds[TOK_PER_BLOCK][NH][NW];

    const int lane = threadIdx.x & 31;
    const int wave = threadIdx.x >> 5;
    const int tIn  = wave * 2 + (lane >> 4);          // token slot in block, 0..15
    const int h    = lane & 15;                        // head, 0..15
    const int gt   = blockIdx.x * TOK_PER_BLOCK + tIn; // global token

    // ---- load x slice (8 wires of this head) and theta -------------------
    const float* xp = x + (size_t)gt * EDIM + h * NW;
    float4 x0 = *(const float4*)(xp);
    float4 x1 = *(const float4*)(xp + 4);
    float4 t0 = *(const float4*)(theta);
    float4 t1 = *(const float4*)(theta + 4);

    float c[NW];
    c[0] = __cosf(x0.x + t0.x);
    c[1] = __cosf(x0.y + t0.y);
    c[2] = __cosf(x0.z + t0.z);
    c[3] = __cosf(x0.w + t0.w);
    c[4] = __cosf(x1.x + t1.x);
    c[5] = __cosf(x1.y + t1.y);
    c[6] = __cosf(x1.z + t1.z);
    c[7] = __cosf(x1.w + t1.w);

    // ---- closed-form qh: prefix products ---------------------------------
    float q[NW];
    float p = c[0];
#pragma unroll
    for (int w = 1; w < NW; ++w) { p *= c[w]; q[w] = p; }
    q[0] = c[1] * c[2] * c[3] * c[4] * c[5] * c[6] * c[7];

#pragma unroll
    for (int w = 0; w < NW; ++w) qh_lds[tIn][h][w] = q[w];
    __syncthreads();

    // ---- scores (16x16 gram of 8-vectors), softmax -----------------------
    const float inv_sqrt_dk = 0.35355339059327379f;   // 1/sqrt(8)
    float sc[NH];
    float mx = -INFINITY;
#pragma unroll
    for (int g = 0; g < NH; ++g) {
        float d = 0.f;
#pragma unroll
        for (int w = 0; w < NW; ++w) d += q[w] * qh_lds[tIn][g][w];
        d *= inv_sqrt_dk;
        sc[g] = d;
        mx = fmaxf(mx, d);
    }
    float denom = 0.f;
#pragma unroll
    for (int g = 0; g < NH; ++g) { sc[g] = __expf(sc[g] - mx); denom += sc[g]; }

    float o[NW];
#pragma unroll
    for (int w = 0; w < NW; ++w) o[w] = 0.f;
#pragma unroll
    for (int g = 0; g < NH; ++g) {
        float a = sc[g];
#pragma unroll
        for (int w = 0; w < NW; ++w) o[w] += a * qh_lds[tIn][g][w];
    }
    float r = 1.f / denom;
#pragma unroll
    for (int w = 0; w < NW; ++w) o[w] *= r;

    // ---- permuted write (swapaxes(1,2).reshape) --------------------------
    // out row' = b*1024 + h*64 + s/16 ; col' = (s%16)*8 + w
    const int b = gt >> 10;
    const int s = gt & 1023;
    const size_t row = (size_t)b * 1024 + (size_t)h * 64 + (s >> 4);
    float* yp = Y + row * EDIM + (s & 15) * NW;
    *(float4*)(yp)     = make_float4(o[0], o[1], o[2], o[3]);
    *(float4*)(yp + 4) = make_float4(o[4], o[5], o[6], o[7]);
}

// ---------------------------------------------------------------------------
// out = Y @ W^T + bias via f32 WMMA (exact precision vs f32 reference).
// One wave computes one 16x16 output tile; K=128 -> 32 x V_WMMA_F32_16X16X4_F32.
// A fragment: lane<16 -> M=lane,K={k0,k0+1}; lane>=16 -> M=lane-16,K={k0+2,k0+3}
//   -> contiguous float2 per lane from Y row.
// B (=W^T) fragment: N = lane&15 on W row n, same K split -> float2 from W row.
// C/D: vgpr r -> row r + 8*(lane>>4), col lane&15.
// ---------------------------------------------------------------------------
__global__ __launch_bounds__(256) void combine_gemm_wmma(
    const float* __restrict__ Y,
    const float* __restrict__ W,
    const float* __restrict__ bias,
    float* __restrict__ out)
{
    const int lane  = threadIdx.x & 31;
    const int wave  = threadIdx.x >> 5;
    const int tileN = wave;              // 8 N-tiles, one per wave
    const int tileM = blockIdx.x;        // 16-row stripe per block
    const int l16   = lane & 15;
    const int kgrp  = (lane >> 4) * 2;   // K sub-offset per lane half

    const float* arow = Y + ((size_t)tileM * 16 + l16) * EDIM + kgrp;
    const float* brow = W + ((size_t)tileN * 16 + l16) * EDIM + kgrp;

    __builtin_prefetch(arow, 0, 0);      // global_prefetch_b8
    __builtin_prefetch(brow, 0, 0);

    v8f acc = {0.f, 0.f, 0.f, 0.f, 0.f, 0.f, 0.f, 0.f};
#pragma unroll
    for (int k0 = 0; k0 < EDIM; k0 += 4) {
        v2f a = *(const v2f*)(arow + k0);
        v2f b = *(const v2f*)(brow + k0);
        acc = __builtin_amdgcn_wmma_f32_16x16x4_f32(
            /*neg_a=*/false, a, /*neg_b=*/false, b,
            /*c_mod=*/(short)0, acc, /*reuse_a=*/false, /*reuse_b=*/false);
    }

    const int n = tileN * 16 + l16;
    const float bcol = bias[n];
    const int rbase = tileM * 16 + 8 * (lane >> 4);
#pragma unroll
    for (int rr = 0; rr < 8; ++rr) {
        out[(size_t)(rbase + rr) * EDIM + n] = acc[rr] + bcol;
    }
}

extern "C" void kernel_launch(void* const* d_in, const int* in_sizes, int n_in,
                              void* d_out, int out_size, void* d_ws, size_t ws_size,
                              hipStream_t stream) {
    const float* x     = (const float*)d_in[0];   // (B,1024,128) f32
    const float* theta = (const float*)d_in[1];   // (8,) f32
    const float* Wc    = (const float*)d_in[2];   // (128,128) f32
    const float* bc    = (const float*)d_in[3];   // (128,) f32
    float* out = (float*)d_out;
    float* Y   = (float*)d_ws;                    // ntok*128 f32 scratch

    const int ntok = in_sizes[0] / EDIM;          // B*S (multiple of 16)

    const int blocks1 = ntok / TOK_PER_BLOCK;
    quantum_attn_kernel<<<blocks1, 256, 0, stream>>>(x, theta, Y);

    const int mtiles = ntok / 16;                 // each block: 1 M-tile x 8 N-tiles
    combine_gemm_wmma<<<mtiles, 256, 0, stream>>>(Y, Wc, bc, out);
}